// DynamicRoIAlignFromVit_43885975830543
// MI455X (gfx1250) — compile-verified
//
#include <hip/hip_runtime.h>
#include <hip/hip_bf16.h>
#include <stdint.h>

// Problem constants (from reference setup_inputs)
#define NB   16      // batch
#define NC   256     // channels
#define NH   100     // feature H
#define NW   152     // feature W
#define NQ   8192    // queries per batch
#define TOPK 100
#define MAXBINS 64   // out_h*out_w (7*7=49) upper bound for LDS sizing

// ---------- CDNA5 async helpers ----------
__device__ __forceinline__ void async_wait0() {
#if __has_builtin(__builtin_amdgcn_s_wait_asynccnt)
  __builtin_amdgcn_s_wait_asynccnt(0);
#else
  asm volatile("s_wait_asynccnt 0x0" ::: "memory");
#endif
}

// Per-lane async copy: 4B global -> LDS (tracked by ASYNCcnt).
// ldsoff is the LDS byte address (low 32 bits of the flat shared pointer).
__device__ __forceinline__ void async_g2lds_b32(unsigned int ldsoff, const void* gaddr) {
  asm volatile("global_load_async_to_lds_b32 %0, %1, off"
               :: "v"(ldsoff), "v"(gaddr) : "memory");
}

// Per-lane async copy: 16B LDS -> global (tracked by ASYNCcnt).
__device__ __forceinline__ void async_lds2g_b128(void* gaddr, unsigned int ldsoff) {
  asm volatile("global_store_async_from_lds_b128 %0, %1, off"
               :: "v"(gaddr), "v"(ldsoff) : "memory");
}

__device__ __forceinline__ unsigned int lds_byte_off(const void* p) {
  // gfx1250 flat shared pointers carry the LDS byte offset in addr[31:0]
  return (unsigned int)(uintptr_t)p;
}

// ---------- Kernel 1: per-batch top-100 -> normalized clipped ROIs ----------
// Key = (monotonic(score) << 32) | (NQ-1-q): max key == max score, ties -> min q
// (matches jax.lax.top_k descending order with stable tie-break).
__global__ __launch_bounds__(1024) void topk_rois_kernel(
    const float* __restrict__ vit,   // [NB, NQ, 6]
    float* __restrict__ rois)        // [NB*TOPK, 4] (xmin,ymin,xmax,ymax) normalized
{
  const int b = blockIdx.x;
  const int t = threadIdx.x;

  __shared__ unsigned int s_key[NQ];                 // 32 KB
  __shared__ unsigned long long s_red[32];
  __shared__ unsigned long long s_win;

  // Stage the strided score column (element 5 of 6) into LDS via the async engine.
  const float* score_base = vit + (size_t)b * NQ * 6 + 5;
  for (int i = t; i < NQ; i += 1024) {
    async_g2lds_b32(lds_byte_off(&s_key[i]), (const void*)(score_base + (size_t)i * 6));
  }
  async_wait0();
  __syncthreads();

  // Monotonic float->uint transform in place.
  for (int i = t; i < NQ; i += 1024) {
    unsigned int u = s_key[i];
    s_key[i] = (u & 0x80000000u) ? ~u : (u | 0x80000000u);
  }
  __syncthreads();

  for (int r = 0; r < TOPK; ++r) {
    // Thread-local max over 8 strided elements.
    unsigned long long best = 0ull;
    for (int i = t; i < NQ; i += 1024) {
      unsigned long long k =
          ((unsigned long long)s_key[i] << 32) | (unsigned int)(NQ - 1 - i);
      if (k > best) best = k;
    }
    // wave32 reduction
    for (int off = 16; off > 0; off >>= 1) {
      unsigned long long o = __shfl_xor(best, off);
      if (o > best) best = o;
    }
    if ((t & 31) == 0) s_red[t >> 5] = best;
    __syncthreads();
    if (t < 32) {
      unsigned long long v = s_red[t];
      for (int off = 16; off > 0; off >>= 1) {
        unsigned long long o = __shfl_xor(v, off);
        if (o > v) v = o;
      }
      if (t == 0) s_win = v;
    }
    __syncthreads();

    const unsigned long long win = s_win;
    const int q = NQ - 1 - (int)(unsigned int)(win & 0xFFFFFFFFu);

    if (t == 0) {
      const float* bx = vit + ((size_t)b * NQ + q) * 6 + 1;
      float x1 = bx[0] * (1.0f / NW), y1 = bx[1] * (1.0f / NH);
      float x2 = bx[2] * (1.0f / NW), y2 = bx[3] * (1.0f / NH);
      float xmn = fminf(fmaxf(fminf(x1, x2), 0.0f), 1.0f);
      float ymn = fminf(fmaxf(fminf(y1, y2), 0.0f), 1.0f);
      float xmx = fminf(fmaxf(fmaxf(x1, x2), 0.0f), 1.0f);
      float ymx = fminf(fmaxf(fmaxf(y1, y2), 0.0f), 1.0f);
      float* o = rois + ((size_t)b * TOPK + r) * 4;
      o[0] = xmn; o[1] = ymn; o[2] = xmx; o[3] = ymx;
      s_key[q] = 0u;  // remove winner for next round
    }
    __syncthreads();
  }
}

// ---------- Kernel 2: ROIAlign (one block per ROI; thread == channel) ----------
__global__ __launch_bounds__(NC) void roialign_kernel(
    const float* __restrict__ feats,  // [NB, NC, NH, NW]
    const float* __restrict__ rois,   // [NB*TOPK, 4] normalized
    const int* __restrict__ ohp, const int* __restrict__ owp,
    float* __restrict__ out)          // [NB*TOPK, NC, oh, ow]
{
  const int n = blockIdx.x;            // ROI index 0..NB*TOPK-1
  const int b = n / TOPK;
  const int c = threadIdx.x;           // channel
  const int oh = *ohp, ow = *owp;
  const int bins = oh * ow;
  if (bins > MAXBINS) return;

  __shared__ int   s_iy0[MAXBINS], s_iy1[MAXBINS], s_ix0[MAXBINS], s_ix1[MAXBINS];
  __shared__ float s_wy1[MAXBINS], s_wx1[MAXBINS];
  __shared__ __align__(16) float s_out[NC * MAXBINS];  // 64 KB staging

  // Per-bin sample geometry (reference math, sampling_ratio = 1).
  if (c < bins) {
    const float* rp = rois + (size_t)n * 4;
    float x1 = rp[0] * (float)NW;
    float y1 = rp[1] * (float)NH;
    float x2 = rp[2] * (float)NW;
    float y2 = rp[3] * (float)NH;
    float roi_w = fmaxf(x2 - x1, 1.0f);
    float roi_h = fmaxf(y2 - y1, 1.0f);
    int ph = c / ow, pw = c % ow;
    float ys = y1 + ((float)ph + 0.5f) * (roi_h / (float)oh);
    float xs = x1 + ((float)pw + 0.5f) * (roi_w / (float)ow);
    ys = fminf(fmaxf(ys, 0.0f), (float)(NH - 1));
    xs = fminf(fmaxf(xs, 0.0f), (float)(NW - 1));
    float y0f = floorf(ys), x0f = floorf(xs);
    int iy0 = (int)y0f, ix0 = (int)x0f;
    s_iy0[c] = iy0;
    s_ix0[c] = ix0;
    s_iy1[c] = min(iy0 + 1, NH - 1);
    s_ix1[c] = min(ix0 + 1, NW - 1);
    s_wy1[c] = ys - y0f;
    s_wx1[c] = xs - x0f;
  }
  __syncthreads();

  // Gather + bilinear: thread owns channel c, loops over bins.
  const float* fc = feats + ((size_t)b * NC + c) * (size_t)(NH * NW);
  for (int j = 0; j < bins; ++j) {
    if (j + 1 < bins) {  // global_prefetch_b8 on next bin's two rows
      __builtin_prefetch(&fc[s_iy0[j + 1] * NW + s_ix0[j + 1]], 0, 0);
      __builtin_prefetch(&fc[s_iy1[j + 1] * NW + s_ix0[j + 1]], 0, 0);
    }
    const int iy0 = s_iy0[j], iy1 = s_iy1[j], ix0 = s_ix0[j], ix1 = s_ix1[j];
    const float wy1 = s_wy1[j], wx1 = s_wx1[j];
    const float wy0 = 1.0f - wy1, wx0 = 1.0f - wx1;
    const float v00 = fc[iy0 * NW + ix0];
    const float v01 = fc[iy0 * NW + ix1];
    const float v10 = fc[iy1 * NW + ix0];
    const float v11 = fc[iy1 * NW + ix1];
    s_out[c * bins + j] = v00 * (wy0 * wx0) + v01 * (wy0 * wx1) +
                          v10 * (wy1 * wx0) + v11 * (wy1 * wx1);
  }
  __syncthreads();

  // Coalesced tile writeout via async LDS -> global engine (16 B per lane).
  float* gbase = out + (size_t)n * NC * bins;
  const int nvec = (NC * bins) >> 2;  // float4 chunks (NC multiple of 4)
  for (int i = c; i < nvec; i += NC) {
    async_lds2g_b128((void*)(gbase + (size_t)i * 4), lds_byte_off(&s_out[i * 4]));
  }
  async_wait0();
}

extern "C" void kernel_launch(void* const* d_in, const int* in_sizes, int n_in,
                              void* d_out, int out_size, void* d_ws, size_t ws_size,
                              hipStream_t stream) {
  const float* feats = (const float*)d_in[0];   // [16,256,100,152]
  const float* vit   = (const float*)d_in[1];   // [16,8192,6]
  const int*   ohp   = (const int*)d_in[2];     // scalar 7
  const int*   owp   = (const int*)d_in[3];     // scalar 7
  float*       out   = (float*)d_out;           // [1600,256,7,7]
  float*       rois  = (float*)d_ws;            // [1600,4] scratch

  topk_rois_kernel<<<NB, 1024, 0, stream>>>(vit, rois);
  roialign_kernel<<<NB * TOPK, NC, 0, stream>>>(feats, rois, ohp, owp, out);
}